// sorter_exact_10222022164545
// MI455X (gfx1250) — compile-verified
//
#include <hip/hip_runtime.h>
#include <hip/hip_bf16.h>

typedef __attribute__((ext_vector_type(2))) float v2f;
typedef __attribute__((ext_vector_type(8))) float v8f;

#define NN    1024
#define BB    64
#define K_SCALE 6.8
#define K_LOG2E 1.44269504088896340736

// ---------------------------------------------------------------------------
// Kernel 1: per-row statistics -> t = SCALE * log2(e) / std
// Closed form: sum_{j>i}(x_j-x_i)^2 = N*Q - S^2 ; sum_{j>i}(x_j-x_i) = W
// with W = sum_j (2j - (N-1)) * x_j ; var = (N*Q - S^2 - W^2/M)/(M-1), M=N^2.
// ---------------------------------------------------------------------------
__global__ __launch_bounds__(256) void row_stats_kernel(const float* __restrict__ x,
                                                        float* __restrict__ tvals) {
    __shared__ float s_s[256], s_q[256], s_w[256];
    const int row = blockIdx.x;
    const int tid = threadIdx.x;
    const float* xr = x + row * NN;

    float s = 0.f, q = 0.f, w = 0.f;
    #pragma unroll
    for (int k = 0; k < NN; k += 256) {
        int j = k + tid;
        float v = xr[j];
        s += v;
        q += v * v;
        w += (float)(2 * j - (NN - 1)) * v;
    }
    s_s[tid] = s; s_q[tid] = q; s_w[tid] = w;
    __syncthreads();
    for (int off = 128; off > 0; off >>= 1) {
        if (tid < off) {
            s_s[tid] += s_s[tid + off];
            s_q[tid] += s_q[tid + off];
            s_w[tid] += s_w[tid + off];
        }
        __syncthreads();
    }
    if (tid == 0) {
        double S = (double)s_s[0], Q = (double)s_q[0], W = (double)s_w[0];
        double M = (double)NN * (double)NN;
        double sumsq = (double)NN * Q - S * S;
        double var = (sumsq - (W * W) / M) / (M - 1.0);
        double stdv = sqrt(var);
        tvals[row] = (float)(K_SCALE * K_LOG2E / stdv);
    }
}

// ---------------------------------------------------------------------------
// Kernel 2: out[i] = (sum_j sigmoid_like + 0.5)/N using exp2-folded sigmoids.
// sigma = 1/(1 + exp2(y_i - y_j)) with y = x * t.
// Each wave owns a 16-row (i) strip; inner loop builds 16x4 f32 A tiles of
// sigma values and accumulates row sums via V_WMMA_F32_16X16X4_F32 with an
// all-ones B matrix: D[m][n] = sum_k A[m][k] + C[m][n].
// A layout (f32 16x4): lane&15 = M; lanes>=16 hold K=2,3; VGPR0=K, VGPR1=K+1.
// ---------------------------------------------------------------------------
__global__ __launch_bounds__(256) void softrank_wmma_kernel(const float* __restrict__ x,
                                                            const float* __restrict__ tvals,
                                                            float* __restrict__ out) {
    __shared__ float y[NN];
    const int row   = blockIdx.x >> 3;   // 8 blocks per row
    const int chunk = blockIdx.x & 7;    // 128-wide i chunk
    const int tid   = threadIdx.x;

    const float t = tvals[row];
    const float* xr = x + row * NN;
    #pragma unroll
    for (int k = tid; k < NN; k += 256) y[k] = xr[k] * t;
    __syncthreads();

    const int lane = tid & 31;
    const int wave = tid >> 5;
    const int i0   = chunk * 128 + wave * 16;
    const int m    = lane & 15;
    const int joff = (lane >> 4) << 1;   // 0 for lanes 0-15, 2 for lanes 16-31
    const float yi = y[i0 + m];

    v8f c0 = {}, c1 = {}, c2 = {}, c3 = {};
    v2f b; b.x = 1.0f; b.y = 1.0f;       // all-ones B (layout-independent)

    for (int j0 = 0; j0 < NN; j0 += 16) {
        // 4 independent sigma pairs -> 4 WMMA accumulator chains (ILP)
        float2 p0 = *reinterpret_cast<const float2*>(&y[j0 +  0 + joff]);
        float2 p1 = *reinterpret_cast<const float2*>(&y[j0 +  4 + joff]);
        float2 p2 = *reinterpret_cast<const float2*>(&y[j0 +  8 + joff]);
        float2 p3 = *reinterpret_cast<const float2*>(&y[j0 + 12 + joff]);

        v2f a0, a1, a2, a3;
        a0.x = __builtin_amdgcn_rcpf(1.0f + __builtin_amdgcn_exp2f(yi - p0.x));
        a0.y = __builtin_amdgcn_rcpf(1.0f + __builtin_amdgcn_exp2f(yi - p0.y));
        a1.x = __builtin_amdgcn_rcpf(1.0f + __builtin_amdgcn_exp2f(yi - p1.x));
        a1.y = __builtin_amdgcn_rcpf(1.0f + __builtin_amdgcn_exp2f(yi - p1.y));
        a2.x = __builtin_amdgcn_rcpf(1.0f + __builtin_amdgcn_exp2f(yi - p2.x));
        a2.y = __builtin_amdgcn_rcpf(1.0f + __builtin_amdgcn_exp2f(yi - p2.y));
        a3.x = __builtin_amdgcn_rcpf(1.0f + __builtin_amdgcn_exp2f(yi - p3.x));
        a3.y = __builtin_amdgcn_rcpf(1.0f + __builtin_amdgcn_exp2f(yi - p3.y));

        c0 = __builtin_amdgcn_wmma_f32_16x16x4_f32(false, a0, false, b, (short)0, c0, false, false);
        c1 = __builtin_amdgcn_wmma_f32_16x16x4_f32(false, a1, false, b, (short)0, c1, false, false);
        c2 = __builtin_amdgcn_wmma_f32_16x16x4_f32(false, a2, false, b, (short)0, c2, false, false);
        c3 = __builtin_amdgcn_wmma_f32_16x16x4_f32(false, a3, false, b, (short)0, c3, false, false);
    }

    // Combine accumulators; every column of D holds the same row sum.
    float r[8];
    #pragma unroll
    for (int i = 0; i < 8; ++i) r[i] = c0[i] + c1[i] + c2[i] + c3[i];

    // C/D layout: VGPR i holds M=i for lanes 0-15, M=8+i for lanes 16-31.
    const float inv_n = 1.0f / (float)NN;
    if (lane == 0) {
        #pragma unroll
        for (int i = 0; i < 8; ++i) out[row * NN + i0 + i] = (r[i] + 0.5f) * inv_n;
    }
    if (lane == 16) {
        #pragma unroll
        for (int i = 0; i < 8; ++i) out[row * NN + i0 + 8 + i] = (r[i] + 0.5f) * inv_n;
    }
}

extern "C" void kernel_launch(void* const* d_in, const int* in_sizes, int n_in,
                              void* d_out, int out_size, void* d_ws, size_t ws_size,
                              hipStream_t stream) {
    (void)in_sizes; (void)n_in; (void)out_size; (void)ws_size;
    const float* x  = (const float*)d_in[0];
    float* out      = (float*)d_out;
    float* tvals    = (float*)d_ws;   // 64 floats of scratch

    row_stats_kernel<<<BB, 256, 0, stream>>>(x, tvals);
    softrank_wmma_kernel<<<BB * 8, 256, 0, stream>>>(x, tvals, out);
}